// GATEncoder_46351287058739
// MI455X (gfx1250) — compile-verified
//
#include <hip/hip_runtime.h>
#include <hip/hip_bf16.h>

// ---------------------------------------------------------------------------
// GAT encoder for MI455X (gfx1250, wave32).
//   layer: h = X@W (WMMA f32 16x16x4), alpha dots, segment softmax over dst,
//          weighted scatter-add, then BN+ReLU between layers.
// ---------------------------------------------------------------------------

#define NNODES 100000
#define NEDGES 1600000
#define EPN    (NEDGES + NNODES)   // edges + self loops = 1,700,000
#define HIDDIM 128
#define NHEAD  4
#define NEG_SLOPE 0.2f

typedef float v2f __attribute__((ext_vector_type(2)));
typedef float v8f __attribute__((ext_vector_type(8)));

// ---- monotonic uint encoding of float for atomicMax-based segment max ------
__device__ __forceinline__ unsigned orderF(float f) {
    unsigned u = __float_as_uint(f);
    return (u & 0x80000000u) ? ~u : (u | 0x80000000u);
}
__device__ __forceinline__ float unorderF(unsigned u) {
    u = (u & 0x80000000u) ? (u & 0x7FFFFFFFu) : ~u;
    return __uint_as_float(u);
}

// ---------------------------------------------------------------------------
// GEMM: C[M,128] = A[M,128] @ B[128,128] using V_WMMA_F32_16X16X4_F32.
// One wave computes one 16x16 output tile; 8 waves/block cover all 8 col
// tiles of a 16-row block. Grid.x = M/16 (M = 100000 divides exactly).
// Lane layouts per CDNA5 ISA 7.12.2 (32-bit A 16x4, 32-bit C/D 16x16).
// ---------------------------------------------------------------------------
__global__ __launch_bounds__(256) void gemm128_wmma(
    const float* __restrict__ A, const float* __restrict__ B,
    float* __restrict__ C)
{
    const int lane = threadIdx.x & 31;
    const int wave = threadIdx.x >> 5;
    const int row0 = blockIdx.x << 4;   // 16 rows per block
    const int col0 = wave << 4;         // 16 cols per wave
    const int r    = lane & 15;         // M (for A) / N (for B,C) within tile
    const int koff = (lane >> 4) << 1;  // K sub-offset: lanes 16-31 take K+2,K+3

    v8f acc = {};
    const float2* A2 = reinterpret_cast<const float2*>(A + (size_t)(row0 + r) * 128);

#pragma unroll
    for (int k = 0; k < 128; k += 4) {
        float2 af = A2[(k + koff) >> 1];           // A[row, k+koff .. k+koff+1]
        v2f a; a.x = af.x; a.y = af.y;
        v2f b;
        b.x = B[(size_t)(k + koff)     * 128 + col0 + r];
        b.y = B[(size_t)(k + koff + 1) * 128 + col0 + r];
        acc = __builtin_amdgcn_wmma_f32_16x16x4_f32(
            /*neg_a=*/false, a, /*neg_b=*/false, b,
            /*c_mod=*/(short)0, acc, /*reuse_a=*/false, /*reuse_b=*/false);
    }

    const int crow = (lane >> 4) << 3;  // lanes 16-31 hold rows M=8..15
#pragma unroll
    for (int i = 0; i < 8; ++i)
        C[(size_t)(row0 + crow + i) * 128 + col0 + r] = acc[i];
}

// ---------------------------------------------------------------------------
// Per-node attention coefficients: as[n,h] = <h[n,h,:], a_src[h,:]>, same dst.
// 32 lanes per node, float4 per lane, 8-lane shfl_xor reduction per head.
// ---------------------------------------------------------------------------
__global__ __launch_bounds__(256) void alpha_kernel(
    const float* __restrict__ h,
    const float* __restrict__ a_src, const float* __restrict__ a_dst,
    float* __restrict__ as, float* __restrict__ ad)
{
    int t    = blockIdx.x * blockDim.x + threadIdx.x;
    int node = t >> 5;
    int lane = t & 31;
    if (node >= NNODES) return;

    float4 hv = reinterpret_cast<const float4*>(h + (size_t)node * 128)[lane];
    float4 sv = reinterpret_cast<const float4*>(a_src)[lane];  // flat [H*C]: lane*4 lands in its head
    float4 dv = reinterpret_cast<const float4*>(a_dst)[lane];

    float s = hv.x * sv.x + hv.y * sv.y + hv.z * sv.z + hv.w * sv.w;
    float d = hv.x * dv.x + hv.y * dv.y + hv.z * dv.z + hv.w * dv.w;
#pragma unroll
    for (int off = 1; off < 8; off <<= 1) {
        s += __shfl_xor(s, off, 32);
        d += __shfl_xor(d, off, 32);
    }
    if ((lane & 7) == 0) {
        int head = lane >> 3;
        as[(size_t)node * NHEAD + head] = s;
        ad[(size_t)node * NHEAD + head] = d;
    }
}

// ---------------------------------------------------------------------------
// Pass 1: per-(edge,head) logit e = leaky_relu(as[src]+ad[dst]); segment max.
// ---------------------------------------------------------------------------
__global__ __launch_bounds__(256) void edge_logits_max(
    const int* __restrict__ ei,
    const float* __restrict__ as, const float* __restrict__ ad,
    float* __restrict__ e_buf, unsigned* __restrict__ m_ord)
{
    int t = blockIdx.x * blockDim.x + threadIdx.x;
    if (t >= EPN * NHEAD) return;
    int e  = t >> 2;
    int hh = t & 3;
    int s, d;
    if (e < NEDGES) { s = ei[e]; d = ei[NEDGES + e]; }
    else            { s = d = e - NEDGES; }          // self loop
    float v = as[(size_t)s * NHEAD + hh] + ad[(size_t)d * NHEAD + hh];
    v = (v > 0.0f) ? v : NEG_SLOPE * v;
    e_buf[t] = v;
    atomicMax(&m_ord[(size_t)d * NHEAD + hh], orderF(v));
}

// ---------------------------------------------------------------------------
// Pass 2: ex = exp(e - m[dst]); segment sum into denom.
// ---------------------------------------------------------------------------
__global__ __launch_bounds__(256) void edge_exp_sum(
    const int* __restrict__ ei,
    float* __restrict__ e_buf, const unsigned* __restrict__ m_ord,
    float* __restrict__ denom)
{
    int t = blockIdx.x * blockDim.x + threadIdx.x;
    if (t >= EPN * NHEAD) return;
    int e  = t >> 2;
    int hh = t & 3;
    int d  = (e < NEDGES) ? ei[NEDGES + e] : (e - NEDGES);
    float m  = unorderF(m_ord[(size_t)d * NHEAD + hh]);
    float ex = __expf(e_buf[t] - m);
    e_buf[t] = ex;
    atomicAdd(&denom[(size_t)d * NHEAD + hh], ex);
}

// ---------------------------------------------------------------------------
// Pass 3: out[dst,:] += (ex/denom[dst]) * h[src,:].  32 lanes/edge, float4
// gather (coalesced 512B per edge), 4 f32 atomics per lane; the 51MB output
// accumulator is L2-resident (192MB L2) so atomics resolve in L2.
// ---------------------------------------------------------------------------
__global__ __launch_bounds__(256) void edge_scatter(
    const int* __restrict__ ei,
    const float* __restrict__ e_buf, const float* __restrict__ denom,
    const float* __restrict__ h, float* __restrict__ out)
{
    int t    = blockIdx.x * blockDim.x + threadIdx.x;
    int e    = t >> 5;
    int lane = t & 31;
    if (e >= EPN) return;
    int s, d;
    if (e < NEDGES) { s = ei[e]; d = ei[NEDGES + e]; }
    else            { s = d = e - NEDGES; }
    int head = lane >> 3;
    float ex  = e_buf[(size_t)e * NHEAD + head];
    float den = denom[(size_t)d * NHEAD + head];
    float coeff = ex / (den + 1e-16f);

    float4 hv = reinterpret_cast<const float4*>(h + (size_t)s * 128)[lane];
    float* o  = out + (size_t)d * 128 + lane * 4;
    atomicAdd(o + 0, coeff * hv.x);
    atomicAdd(o + 1, coeff * hv.y);
    atomicAdd(o + 2, coeff * hv.z);
    atomicAdd(o + 3, coeff * hv.w);
}

// ---------------------------------------------------------------------------
// BN stats: add bias in-place, accumulate per-channel sum / sumsq.
// blockDim = 128 (thread == channel), 256 rows per block.
// ---------------------------------------------------------------------------
__global__ __launch_bounds__(128) void bn_stats(
    float* __restrict__ agg, const float* __restrict__ bias,
    float* __restrict__ sums /* [0..127]=sum, [128..255]=sumsq */)
{
    int c    = threadIdx.x;
    int base = blockIdx.x * 256;
    int end  = min(base + 256, NNODES);
    float bc = bias[c];
    float s = 0.0f, ss = 0.0f;
    for (int n = base; n < end; ++n) {
        float v = agg[(size_t)n * 128 + c] + bc;
        agg[(size_t)n * 128 + c] = v;
        s  += v;
        ss += v * v;
    }
    atomicAdd(&sums[c], s);
    atomicAdd(&sums[128 + c], ss);
}

__global__ __launch_bounds__(256) void bn_norm_relu(
    float* __restrict__ agg, const float* __restrict__ sums,
    const float* __restrict__ gamma, const float* __restrict__ beta)
{
    int idx = blockIdx.x * blockDim.x + threadIdx.x;
    if (idx >= NNODES * 128) return;
    int c = idx & 127;
    const float invN = 1.0f / (float)NNODES;
    float mean = sums[c] * invN;
    float var  = sums[128 + c] * invN - mean * mean;
    float v = gamma[c] * (agg[idx] - mean) * rsqrtf(var + 1e-5f) + beta[c];
    agg[idx] = (v > 0.0f) ? v : 0.0f;
}

__global__ __launch_bounds__(256) void bias_add(
    float* __restrict__ out, const float* __restrict__ b)
{
    int idx = blockIdx.x * blockDim.x + threadIdx.x;
    if (idx >= NNODES * 128) return;
    out[idx] += b[idx & 127];
}

// ---------------------------------------------------------------------------
extern "C" void kernel_launch(void* const* d_in, const int* in_sizes, int n_in,
                              void* d_out, int out_size, void* d_ws, size_t ws_size,
                              hipStream_t stream)
{
    (void)in_sizes; (void)n_in; (void)ws_size;
    const float* x      = (const float*)d_in[0];
    const int*   ei     = (const int*)  d_in[1];   // [2,E]: src row 0, dst row 1
    const float* W1     = (const float*)d_in[2];
    const float* a_src1 = (const float*)d_in[3];
    const float* a_dst1 = (const float*)d_in[4];
    const float* b1     = (const float*)d_in[5];
    const float* gamma1 = (const float*)d_in[6];
    const float* beta1  = (const float*)d_in[7];
    const float* W2     = (const float*)d_in[8];
    const float* a_src2 = (const float*)d_in[9];
    const float* a_dst2 = (const float*)d_in[10];
    const float* b2     = (const float*)d_in[11];
    float* out = (float*)d_out;

    // ---- workspace carving -------------------------------------------------
    char* ws = (char*)d_ws;
    size_t off = 0;
    auto alloc = [&](size_t bytes) -> void* {
        void* p = ws + off;
        off += (bytes + 255) & ~(size_t)255;
        return p;
    };
    float*    h     = (float*)   alloc((size_t)NNODES * 128 * 4);  // GEMM output
    float*    agg   = (float*)   alloc((size_t)NNODES * 128 * 4);  // layer-1 aggregate / layer-2 input
    float*    e_buf = (float*)   alloc((size_t)EPN * NHEAD * 4);   // per-edge logits -> exp
    float*    as    = (float*)   alloc((size_t)NNODES * NHEAD * 4);
    float*    ad    = (float*)   alloc((size_t)NNODES * NHEAD * 4);
    unsigned* m_ord = (unsigned*)alloc((size_t)NNODES * NHEAD * 4);
    float*    denom = (float*)   alloc((size_t)NNODES * NHEAD * 4);
    float*    sums  = (float*)   alloc(256 * 4);

    const int gGemm    = NNODES / 16;                       // 6250, exact
    const int gAlpha   = (NNODES * 32 + 255) / 256;
    const int gEdgeH   = (EPN * NHEAD + 255) / 256;
    const int gScatter = (EPN * 32 + 255) / 256;
    const int gElem    = (NNODES * 128 + 255) / 256;

    // ===================== layer 1 =====================
    hipMemsetAsync(agg,   0, (size_t)NNODES * 128 * 4, stream);
    hipMemsetAsync(m_ord, 0, (size_t)NNODES * NHEAD * 4, stream); // orderF(-inf)-like floor
    hipMemsetAsync(denom, 0, (size_t)NNODES * NHEAD * 4, stream);
    hipMemsetAsync(sums,  0, 256 * 4, stream);
    hipMemsetAsync(out,   0, (size_t)out_size * 4, stream);

    gemm128_wmma<<<gGemm, 256, 0, stream>>>(x, W1, h);
    alpha_kernel<<<gAlpha, 256, 0, stream>>>(h, a_src1, a_dst1, as, ad);
    edge_logits_max<<<gEdgeH, 256, 0, stream>>>(ei, as, ad, e_buf, m_ord);
    edge_exp_sum<<<gEdgeH, 256, 0, stream>>>(ei, e_buf, m_ord, denom);
    edge_scatter<<<gScatter, 256, 0, stream>>>(ei, e_buf, denom, h, agg);
    bn_stats<<<(NNODES + 255) / 256, 128, 0, stream>>>(agg, b1, sums);
    bn_norm_relu<<<gElem, 256, 0, stream>>>(agg, sums, gamma1, beta1);

    // ===================== layer 2 =====================
    hipMemsetAsync(m_ord, 0, (size_t)NNODES * NHEAD * 4, stream);
    hipMemsetAsync(denom, 0, (size_t)NNODES * NHEAD * 4, stream);

    gemm128_wmma<<<gGemm, 256, 0, stream>>>(agg, W2, h);
    alpha_kernel<<<gAlpha, 256, 0, stream>>>(h, a_src2, a_dst2, as, ad);
    edge_logits_max<<<gEdgeH, 256, 0, stream>>>(ei, as, ad, e_buf, m_ord);
    edge_exp_sum<<<gEdgeH, 256, 0, stream>>>(ei, e_buf, m_ord, denom);
    edge_scatter<<<gScatter, 256, 0, stream>>>(ei, e_buf, denom, h, out);
    bias_add<<<gElem, 256, 0, stream>>>(out, b2);
}